// CoLightNet_46351287059074
// MI455X (gfx1250) — compile-verified
//
#include <hip/hip_runtime.h>

// CoLightNet forward for MI455X (gfx1250, wave32).
// All matrix math on V_WMMA_F32_16X16X4_F32 (exact fp32, matches fp32 reference).
// Attention is single-pass flash-style: the 256MB adjacency matrix (the roofline
// limiter at 23.3 TB/s) is streamed exactly once, with explicit prefetch.

typedef float v2f __attribute__((ext_vector_type(2)));
typedef float v8f __attribute__((ext_vector_type(8)));

__device__ __forceinline__ v8f wmma4(v2f a, v2f b, v8f c) {
  // D(16x16,f32) = A(16x4,f32) * B(4x16,f32) + C
  return __builtin_amdgcn_wmma_f32_16x16x4_f32(false, a, false, b, (short)0, c,
                                               false, false);
}

// ---------------------------------------------------------------------------
// GEMM: C[M,NCOLS] = act(A[M,128] @ B[128,NCOLS] + bias[NCOLS]).
// One wave per 16x16 output tile. NCOLS is a compile-time constant so the
// column guard vanishes for NCOLS=128 and all K-offsets are immediates.
// ---------------------------------------------------------------------------
template <int NCOLS, int RELU>
__global__ __launch_bounds__(256) void gemm_bias_act(
    const float* __restrict__ A, const float* __restrict__ B,
    const float* __restrict__ bias, float* __restrict__ C, int M)
{
  constexpr int K = 128;
  const int lane = threadIdx.x & 31;
  const int wave = threadIdx.x >> 5;
  const int half = lane >> 4;   // 0: K-pair {0,1} / rows 0..7   1: {2,3} / rows 8..15
  const int l16  = lane & 15;

  constexpr int colTiles = (NCOLS + 15) / 16;
  const int rowTiles = M >> 4;
  const int tile = blockIdx.x * 8 + wave;
  if (tile >= rowTiles * colTiles) return;   // wave-uniform (EXEC stays full for WMMA)
  const int tm = tile / colTiles;
  const int tn = tile - tm * colTiles;

  const int arow = (tm << 4) + l16;          // A-operand: lane = M
  const int col  = (tn << 4) + l16;          // B/C: lane = N
  const bool cok = (NCOLS % 16 == 0) || (col < NCOLS);   // folds away for NCOLS=128

  const float* Arow = A + (size_t)arow * K;
  const float* Bcol = B + (cok ? col : 0);   // safe base even for guarded lanes

  v8f acc = {};
  #pragma unroll 8
  for (int kk = 0; kk < K; kk += 4) {
    const int kb = kk + half * 2;
    v2f a = *(const v2f*)(Arow + kb);                    // A[m][kb], A[m][kb+1]
    v2f b;
    b[0] = cok ? Bcol[(size_t)kb * NCOLS]       : 0.0f;  // B[kb][n]
    b[1] = cok ? Bcol[(size_t)(kb + 1) * NCOLS] : 0.0f;  // B[kb+1][n]
    acc = wmma4(a, b, acc);
  }

  const float bv = (cok && bias) ? bias[col] : 0.0f;
  float* Cb = C + (size_t)((tm << 4) + half * 8) * NCOLS + (cok ? col : 0);
  #pragma unroll
  for (int r = 0; r < 8; ++r) {
    float v = acc[r] + bv;
    if (RELU) v = fmaxf(v, 0.0f);
    if (cok) Cb[(size_t)r * NCOLS] = v;      // C layout: VGPR r -> rows r / r+8
  }
}

// ---------------------------------------------------------------------------
// Flash attention with adjacency mask.
// Block = 4 waves = 128 threads, one 16-row query tile per block.
// Each wave owns key tiles j = wave, wave+4, ... with private online-softmax
// state; the 4 partials merge through LDS at the end.
// ---------------------------------------------------------------------------
#define NN 8192
#define EE 128
#define KT 512   // NN/16 key tiles
#define AW 4     // waves per block

__device__ __forceinline__ float half_max16(float v) {
  #pragma unroll
  for (int o = 1; o < 16; o <<= 1) v = fmaxf(v, __shfl_xor(v, o, 32));
  return v;
}
__device__ __forceinline__ float half_sum16(float v) {
  #pragma unroll
  for (int o = 1; o < 16; o <<= 1) v += __shfl_xor(v, o, 32);
  return v;
}

__global__ __launch_bounds__(128) void attn_flash(
    const float* __restrict__ Q, const float* __restrict__ Km,
    const float* __restrict__ H, const int* __restrict__ adj,
    float* __restrict__ Agg)
{
  __shared__ float lds_p[AW][16 * 16];     // per-wave P tile (C-layout -> A-layout)
  __shared__ float lds_agg[AW][16 * EE];   // per-wave partial aggregates
  __shared__ float lds_m[AW][16];
  __shared__ float lds_l[AW][16];

  const int lane = threadIdx.x & 31;
  const int wave = threadIdx.x >> 5;
  const int half = lane >> 4;
  const int l16  = lane & 15;
  const int qrow = blockIdx.x << 4;

  float mrow[8], lrow[8];
  v8f agg[8];                              // agg[t][r] = row(half*8+r), col(t*16+l16)
  v8f vzero = {};
  #pragma unroll
  for (int r = 0; r < 8; ++r) { mrow[r] = -1e30f; lrow[r] = 0.0f; }
  #pragma unroll
  for (int t = 0; t < 8; ++t) agg[t] = vzero;

  const float scale = 0.088388347648318447f;  // 1/sqrt(128)
  const float* Qrow = Q + (size_t)(qrow + l16) * EE;

  for (int j = wave; j < KT; j += AW) {       // 128 uniform iterations per wave
    const int kcol = j << 4;

    // Prefetch next key tile's adjacency + K rows (hides the 256MB adj stream).
    if (j + AW < KT) {
      const int nk = (j + AW) << 4;
      __builtin_prefetch(&adj[(size_t)(qrow + half * 8) * NN + nk + l16], 0, 1);
      __builtin_prefetch(&Km[(size_t)(nk + l16) * EE + half * 64], 0, 1);
    }

    // ---- S = Q_tile @ K_tileᵀ : B-operand B[k][n] = Km[kcol+n][k] (contiguous v2f)
    v8f s = {};
    const float* Krow = Km + (size_t)(kcol + l16) * EE;
    #pragma unroll 8
    for (int kk = 0; kk < EE; kk += 4) {
      const int kb = kk + half * 2;
      v2f a = *(const v2f*)(Qrow + kb);
      v2f b = *(const v2f*)(Krow + kb);
      s = wmma4(a, b, s);
    }

    // ---- mask + online softmax update (rows split across half-waves)
    #pragma unroll
    for (int r = 0; r < 8; ++r) {
      const int grow = qrow + half * 8 + r;
      const int av   = adj[(size_t)grow * NN + (kcol + l16)];
      const float sv = (av != 0) ? (s[r] * scale) : -1e30f;
      const float mx   = half_max16(sv);
      const float mnew = fmaxf(mrow[r], mx);
      const float corr = __expf(mrow[r] - mnew);
      const float p    = (av != 0) ? __expf(sv - mnew) : 0.0f;
      lrow[r] = lrow[r] * corr + half_sum16(p);
      mrow[r] = mnew;
      #pragma unroll
      for (int t = 0; t < 8; ++t) agg[t][r] *= corr;
      lds_p[wave][(half * 8 + r) * 16 + l16] = p;   // row-major 16x16 stage
    }
    // Per-wave DS ops are in-order in HW; just stop compiler reordering.
    __asm__ volatile("" ::: "memory");

    // ---- reload P in A-operand layout (lane = M, half selects K pair)
    v2f pa[4];
    #pragma unroll
    for (int ki = 0; ki < 4; ++ki) {
      const int kb = ki * 4 + half * 2;
      pa[ki][0] = lds_p[wave][l16 * 16 + kb];
      pa[ki][1] = lds_p[wave][l16 * 16 + kb + 1];
    }

    // ---- agg += P(16x16) @ H[kcol:kcol+16][0:128]
    const float* Hb = H + (size_t)kcol * EE + l16;
    #pragma unroll
    for (int t = 0; t < 8; ++t) {
      v8f acc = agg[t];
      #pragma unroll
      for (int ki = 0; ki < 4; ++ki) {
        const int kb = ki * 4 + half * 2;
        v2f b;
        b[0] = Hb[(size_t)kb * EE       + t * 16];
        b[1] = Hb[(size_t)(kb + 1) * EE + t * 16];
        acc = wmma4(pa[ki], b, acc);
      }
      agg[t] = acc;
    }
  }

  // ---- spill per-wave partials, merge 4 waves
  #pragma unroll
  for (int r = 0; r < 8; ++r) {
    lds_m[wave][half * 8 + r] = mrow[r];
    lds_l[wave][half * 8 + r] = lrow[r];
  }
  #pragma unroll
  for (int t = 0; t < 8; ++t)
    #pragma unroll
    for (int r = 0; r < 8; ++r)
      lds_agg[wave][(half * 8 + r) * EE + t * 16 + l16] = agg[t][r];
  __syncthreads();

  for (int e = threadIdx.x; e < 16 * EE; e += 128) {
    const int r = e >> 7;
    const int c = e & 127;
    float m = -1e30f;
    #pragma unroll
    for (int w = 0; w < AW; ++w) m = fmaxf(m, lds_m[w][r]);
    float l = 0.0f, a = 0.0f;
    #pragma unroll
    for (int w = 0; w < AW; ++w) {
      const float f = __expf(lds_m[w][r] - m);
      l += lds_l[w][r] * f;
      a += lds_agg[w][r * EE + c] * f;
    }
    Agg[(size_t)(qrow + r) * EE + c] = a / l;
  }
}

// ---------------------------------------------------------------------------
extern "C" void kernel_launch(void* const* d_in, const int* in_sizes, int n_in,
                              void* d_out, int out_size, void* d_ws, size_t ws_size,
                              hipStream_t stream) {
  (void)in_sizes; (void)n_in; (void)out_size; (void)ws_size;
  const float* X   = (const float*)d_in[0];
  const int*   adj = (const int*)  d_in[1];
  const float* w1  = (const float*)d_in[2];
  const float* b1  = (const float*)d_in[3];
  const float* w2  = (const float*)d_in[4];
  const float* b2  = (const float*)d_in[5];
  const float* wq  = (const float*)d_in[6];
  const float* wk  = (const float*)d_in[7];
  const float* wh1 = (const float*)d_in[8];
  const float* bh1 = (const float*)d_in[9];
  const float* wh2 = (const float*)d_in[10];
  const float* bh2 = (const float*)d_in[11];
  float* out = (float*)d_out;

  const int Nn = 8192;
  const size_t NE = (size_t)Nn * EE;
  float* t   = (float*)d_ws;   // 4MB each; 20MB total workspace use
  float* h   = t + NE;
  float* q   = h + NE;
  float* k   = q + NE;
  float* agg = k + NE;

  auto blocksFor = [](int M, int N) {
    int tiles = (M >> 4) * ((N + 15) >> 4);
    return (tiles + 7) / 8;                // 8 waves (tiles) per 256-thread block
  };
  const int g128 = blocksFor(Nn, EE);
  const int g8   = blocksFor(Nn, 8);

  // state_mlp
  gemm_bias_act<128, 1><<<g128, 256, 0, stream>>>(X,   w1,  b1,      t,   Nn);
  gemm_bias_act<128, 0><<<g128, 256, 0, stream>>>(t,   w2,  b2,      h,   Nn);
  // q/k projections (no bias)
  gemm_bias_act<128, 0><<<g128, 256, 0, stream>>>(h,   wq,  nullptr, q,   Nn);
  gemm_bias_act<128, 0><<<g128, 256, 0, stream>>>(h,   wk,  nullptr, k,   Nn);
  // masked softmax attention + aggregation (adj streamed once)
  attn_flash<<<Nn / 16, 128, 0, stream>>>(q, k, h, adj, agg);
  // output head
  gemm_bias_act<128, 1><<<g128, 256, 0, stream>>>(agg, wh1, bh1,     t,   Nn);
  gemm_bias_act<8,   0><<<g8,   256, 0, stream>>>(t,   wh2, bh2,     out, Nn);
}